// DataModule_42262478192914
// MI455X (gfx1250) — compile-verified
//
#include <hip/hip_runtime.h>
#include <cstdint>

// proc_pos : [16, 8]   f32  (512 B  -> LDS via async global->LDS)
// locs_sp  : [500000,3] f32 (6 MB   -> L2-resident gather)
// ids      : [8M] int       (coalesced streams, prefetched)
// out      : [8M, 11] f32   (352 MB -> NT b128 streaming stores)
#define N_PROCS         16
#define PROC_DIM        8
#define SPATIAL_DIM     3
#define ROW_DIM         11
#define ROWS_PER_BLOCK  256   // 1 row per thread; tile = 256*44B = 11264 B (16-divisible)

typedef float v4f __attribute__((ext_vector_type(4)));
typedef int   v4i __attribute__((ext_vector_type(4)));
typedef __attribute__((address_space(1))) v4i gv4i_t;  // global (AS1)
typedef __attribute__((address_space(3))) v4i lv4i_t;  // LDS    (AS3)

__global__ __launch_bounds__(ROWS_PER_BLOCK) void gather_concat_kernel(
    const float* __restrict__ proc_pos,
    const float* __restrict__ locs_sp,
    const int*   __restrict__ process_ids,
    const int*   __restrict__ location_ids,
    float*       __restrict__ out,
    unsigned     nRows)
{
    __shared__ __align__(16) float sProc[N_PROCS * PROC_DIM];        // 512 B
    __shared__ __align__(16) float sTile[ROWS_PER_BLOCK * ROW_DIM];  // 11264 B

    const unsigned tid = threadIdx.x;

    // ---- Stage proc_pos into LDS via the CDNA5 async path (ASYNCcnt).
#if defined(__gfx1250__) && __has_builtin(__builtin_amdgcn_global_load_async_to_lds_b128)
    if (tid < 32u) {   // exactly wave 0: wave-uniform branch
        const char* gsrc = (const char*)proc_pos + tid * 16u;
        char*       ldst = (char*)sProc          + tid * 16u;
        __builtin_amdgcn_global_load_async_to_lds_b128(
            (gv4i_t*)(uintptr_t)gsrc,
            (lv4i_t*)(uint32_t)(uintptr_t)ldst,
            /*offset=*/0, /*cpol=*/0);
#if __has_builtin(__builtin_amdgcn_s_wait_asynccnt)
        __builtin_amdgcn_s_wait_asynccnt(0);
#else
        asm volatile("s_wait_asynccnt 0" ::: "memory");
#endif
    }
    // Opaque never-true store: the async builtin only saw a laundered LDS
    // pointer, so without a visible store the optimizer may fold sProc loads.
    {
        unsigned never = 0u;
        asm volatile("" : "+s"(never));
        if (tid < never)
            sProc[tid & 127u] = proc_pos[tid & 127u];
    }
#else
    if (tid < (unsigned)(N_PROCS * PROC_DIM))
        sProc[tid] = proc_pos[tid];
#endif
    __syncthreads();

    const unsigned rowBase = blockIdx.x * ROWS_PER_BLOCK;
    const unsigned row     = rowBase + tid;
    unsigned R = nRows - rowBase;                 // rows this block owns
    if (R > ROWS_PER_BLOCK) R = ROWS_PER_BLOCK;

    // ---- Phase A: one row per thread, branch-free gather into the LDS tile.
    if (row < nRows) {
        if ((tid & 31u) == 0u) {   // one prefetch per wave (global_prefetch_b8)
            __builtin_prefetch(process_ids  + row + 4096, 0, 1);
            __builtin_prefetch(location_ids + row + 4096, 0, 1);
        }
        const int pid = process_ids[row];   // coalesced b32 loads
        const int lid = location_ids[row];

        const v4f* pp = (const v4f*)(sProc + (pid << 3));  // 32B-aligned
        const v4f  p0 = pp[0];                             // ds_load_b128
        const v4f  p1 = pp[1];                             // ds_load_b128

        const float* lp = locs_sp + lid * SPATIAL_DIM;     // L2-resident gather
        const float  l0 = lp[0], l1 = lp[1], l2 = lp[2];

        float* t = sTile + tid * ROW_DIM;
        t[0] = p0.x; t[1] = p0.y; t[2]  = p0.z; t[3] = p0.w;
        t[4] = p1.x; t[5] = p1.y; t[6]  = p1.z; t[7] = p1.w;
        t[8] = l0;   t[9] = l1;   t[10] = l2;
        // bank check: addr/4 = 11*tid + j, 11 coprime to 64 -> conflict-free
    }
    __syncthreads();

    // ---- Phase B: stream the tile to global, perfectly coalesced NT b128.
    // Block's output window starts at rowBase*44 bytes = blockIdx*11264 B,
    // which is 16B-aligned.
    const unsigned nFloats = R * ROW_DIM;
    const unsigned nF4     = nFloats >> 2;
    v4f*       out4  = (v4f*)(out + (size_t)rowBase * ROW_DIM);
    const v4f* tile4 = (const v4f*)sTile;
    for (unsigned i = tid; i < nF4; i += ROWS_PER_BLOCK) {
        v4f r = tile4[i];                       // ds_load_b128
#if __has_builtin(__builtin_nontemporal_store)
        __builtin_nontemporal_store(r, &out4[i]);  // global_store_b128 NT
#else
        out4[i] = r;
#endif
    }
    const unsigned rem = nFloats & 3u;          // 0 for full blocks
    if (tid < rem) {
        out[(size_t)rowBase * ROW_DIM + nF4 * 4u + tid] = sTile[nF4 * 4u + tid];
    }
}

extern "C" void kernel_launch(void* const* d_in, const int* in_sizes, int n_in,
                              void* d_out, int out_size, void* d_ws, size_t ws_size,
                              hipStream_t stream) {
    (void)in_sizes; (void)n_in; (void)d_ws; (void)ws_size;

    const float* proc_pos     = (const float*)d_in[0];
    const float* locs_sp      = (const float*)d_in[1];
    const int*   process_ids  = (const int*)d_in[2];
    const int*   location_ids = (const int*)d_in[3];
    float*       out          = (float*)d_out;

    const unsigned nRows = (unsigned)(out_size / ROW_DIM);   // 8,000,000
    if (nRows == 0u) return;
    dim3 block(ROWS_PER_BLOCK);
    dim3 grid((nRows + ROWS_PER_BLOCK - 1u) / ROWS_PER_BLOCK);  // 31,250
    gather_concat_kernel<<<grid, block, 0, stream>>>(
        proc_pos, locs_sp, process_ids, location_ids, out, nRows);
}